// SSIM_35210141893401
// MI455X (gfx1250) — compile-verified
//
#include <hip/hip_runtime.h>
#include <cstdint>
#include <cstddef>

// SSIM map, B=16 C=3 H=384 W=640 f32. Memory-bound: ~142MB traffic -> ~6.1us
// floor at 23.3 TB/s. Strategy: async global->LDS halo tiles (CDNA5 async
// path, ASYNCcnt), separable 3x3 box sums from LDS, nontemporal stores.

#define TILE_H 16
#define TILE_W 128
#define IN_H   (TILE_H + 2)    // 18 rows incl. halo
#define IN_W   (TILE_W + 2)    // 130 cols incl. halo
#define LDS_W  (TILE_W + 4)    // 132: padded stride, conflict-free

#if defined(__HIP_DEVICE_COMPILE__) && \
    __has_builtin(__builtin_amdgcn_global_load_async_to_lds_b32) && \
    __has_builtin(__builtin_amdgcn_s_wait_asynccnt)
#define USE_ASYNC_LDS 1
#else
#define USE_ASYNC_LDS 0
#endif

typedef __attribute__((address_space(1))) int GlobInt;   // global (AS1)
typedef __attribute__((address_space(3))) int LdsInt;    // LDS (AS3)

__device__ __forceinline__ int reflect_idx(int i, int n) {
  i = (i < 0) ? -i : i;                // pad=1: -1 -> 1
  i = (i >= n) ? (2 * n - 2 - i) : i;  // n -> n-2
  return i;
}

__global__ void __launch_bounds__(256)
ssim_tile_kernel(const float* __restrict__ x, const float* __restrict__ y,
                 float* __restrict__ out) {
  constexpr int H = 384, W = 640;
  constexpr int TILES_X = W / TILE_W;   // 5
  constexpr int TILES_Y = H / TILE_H;   // 24

  __shared__ float sx[IN_H * LDS_W];
  __shared__ float sy[IN_H * LDS_W];

  const int tid = threadIdx.x;
  const int bid = blockIdx.x;
  const int plane = bid / (TILES_X * TILES_Y);
  const int rem   = bid - plane * (TILES_X * TILES_Y);
  const int ty    = rem / TILES_X;
  const int tx    = rem - ty * TILES_X;
  const int row0  = ty * TILE_H;
  const int col0  = tx * TILE_W;

  const size_t pbase = (size_t)plane * (size_t)(H * W);
  const float* __restrict__ xp = x + pbase;
  const float* __restrict__ yp = y + pbase;

  // ---- Stage halo tiles of x and y into LDS (async global->LDS path) ----
  for (int i = tid; i < IN_H * IN_W; i += 256) {
    const int lr = i / IN_W;
    const int lc = i - lr * IN_W;
    const int gr = reflect_idx(row0 - 1 + lr, H);
    const int gc = reflect_idx(col0 - 1 + lc, W);
    const int g  = gr * W + gc;
    const int l  = lr * LDS_W + lc;
#if USE_ASYNC_LDS
    __builtin_amdgcn_global_load_async_to_lds_b32(
        (GlobInt*)(xp + g), (LdsInt*)(sx + l), 0, 0);
    __builtin_amdgcn_global_load_async_to_lds_b32(
        (GlobInt*)(yp + g), (LdsInt*)(sy + l), 0, 0);
#else
    sx[l] = __builtin_nontemporal_load(xp + g);
    sy[l] = __builtin_nontemporal_load(yp + g);
#endif
  }
#if USE_ASYNC_LDS
  __builtin_amdgcn_s_wait_asynccnt(0);
#endif
  __syncthreads();

  // ---- Separable 3x3 box sums: one column per thread, 8 rows each ----
  const int c     = tid & (TILE_W - 1);     // local column 0..127
  const int half  = tid >> 7;               // 0 or 1
  const int rbase = half * (TILE_H / 2);    // 0 or 8

  const float C1f = (float)(0.01 * 0.01);
  const float C2f = (float)(0.03 * 0.03);
  const float inv9 = 1.0f / 9.0f;

  float hx[3], hy[3], hxx[3], hyy[3], hxy[3];  // 3-row ring of horiz. sums

#pragma unroll
  for (int j = 0; j < (TILE_H / 2) + 2; ++j) {   // 10 input rows of h-sums
    const int lr = rbase + j;
    const float* __restrict__ rx = &sx[lr * LDS_W + c];
    const float* __restrict__ ry = &sy[lr * LDS_W + c];
    const float a0 = rx[0], a1 = rx[1], a2 = rx[2];
    const float b0 = ry[0], b1 = ry[1], b2 = ry[2];
    const int s = j % 3;
    hx[s]  = a0 + a1 + a2;
    hy[s]  = b0 + b1 + b2;
    hxx[s] = a0 * a0 + a1 * a1 + a2 * a2;
    hyy[s] = b0 * b0 + b1 * b1 + b2 * b2;
    hxy[s] = a0 * b0 + a1 * b1 + a2 * b2;

    if (j >= 2) {
      const float mu_x = (hx[0] + hx[1] + hx[2]) * inv9;
      const float mu_y = (hy[0] + hy[1] + hy[2]) * inv9;
      const float exx  = (hxx[0] + hxx[1] + hxx[2]) * inv9;
      const float eyy  = (hyy[0] + hyy[1] + hyy[2]) * inv9;
      const float exy  = (hxy[0] + hxy[1] + hxy[2]) * inv9;

      const float sig_x  = exx - mu_x * mu_x;
      const float sig_y  = eyy - mu_y * mu_y;
      const float sig_xy = exy - mu_x * mu_y;

      const float n = (2.0f * mu_x * mu_y + C1f) * (2.0f * sig_xy + C2f);
      const float d = (mu_x * mu_x + mu_y * mu_y + C1f) *
                      (sig_x + sig_y + C2f);
      float v = (1.0f - n / d) * 0.5f;
      v = fminf(fmaxf(v, 0.0f), 1.0f);

      const int r = rbase + (j - 2);   // local output row
      __builtin_nontemporal_store(
          v, out + pbase + (size_t)(row0 + r) * W + (col0 + c));
    }
  }
}

extern "C" void kernel_launch(void* const* d_in, const int* in_sizes, int n_in,
                              void* d_out, int out_size, void* d_ws, size_t ws_size,
                              hipStream_t stream) {
  (void)n_in; (void)out_size; (void)d_ws; (void)ws_size;
  const float* x = (const float*)d_in[0];
  const float* y = (const float*)d_in[1];
  float* out = (float*)d_out;

  constexpr int H = 384, W = 640;
  const int planes = in_sizes[0] / (H * W);                 // 48
  const int blocks = planes * (H / TILE_H) * (W / TILE_W);  // 5760

  hipLaunchKernelGGL(ssim_tile_kernel, dim3(blocks), dim3(256), 0, stream,
                     x, y, out);
}